// ScalarMSE_66786741453355
// MI455X (gfx1250) — compile-verified
//
#include <hip/hip_runtime.h>

typedef float v2f __attribute__((ext_vector_type(2)));
typedef float v8f __attribute__((ext_vector_type(8)));

#define WAVES_PER_BLOCK 8
#define BLOCK_THREADS (WAVES_PER_BLOCK * 32)
#define MAX_N 12288

// BMC loss: loss[i] = 2*nv*ln2*log2( sum_j 2^(c*p_i*g_j + a_i + b_j) ) + (p_i-g_i)^2
// where the 16x16 tile of base-2 logits is produced by ONE v_wmma_f32_16x16x4_f32:
//   A row i = (c*p_i, a_i, 1, 0),  B col j = (g_j, 1, b_j, 0)
__global__ __launch_bounds__(BLOCK_THREADS)
void bmc_loss_wmma_kernel(const float* __restrict__ pred,
                          const float* __restrict__ gt,
                          const float* __restrict__ sigma,
                          float* __restrict__ out,
                          int N) {
    __shared__ float gs[MAX_N];

    const int tid = threadIdx.x;
    for (int i = tid; i < N; i += BLOCK_THREADS) gs[i] = gt[i];
    __syncthreads();

    const int lane    = tid & 31;
    const int wave    = tid >> 5;
    const int rowTile = blockIdx.x * WAVES_PER_BLOCK + wave;
    const int numTiles = N >> 4;                  // N/16 tiles in each dim
    if (rowTile >= numTiles) return;              // wave-uniform branch (EXEC stays all-1)
    const int rbase = rowTile << 4;

    const float s      = sigma[0];
    const float nv     = s * s;
    const float inv_nv = 1.0f / nv;
    const float LOG2E  = 1.44269504088896340736f;
    const float cA     = LOG2E * inv_nv;          // scales p_i*g_j
    const float kb     = -0.5f * LOG2E * inv_nv;  // a_i = kb*p_i^2, b_j = kb*g_j^2

    // A fragment (16x4 f32): lanes 0-15 hold row M=lane, K=0,1; lanes 16-31 hold K=2,3.
    const int  rl = lane & 15;
    const bool lo = lane < 16;
    const float pi = pred[rbase + rl];

    v2f A;
    A.x = lo ? (pi * cA)      : 1.0f;   // K0 | K2
    A.y = lo ? (kb * pi * pi) : 0.0f;   // K1 | K3
    const float B1 = lo ? 1.0f : 0.0f;  // B K1 | K3 (loop-invariant)

    v8f acc = {0.f,0.f,0.f,0.f,0.f,0.f,0.f,0.f};

    #pragma unroll 4
    for (int J = 0; J < numTiles; ++J) {
        // B fragment (4x16 f32): lanes 0-15 hold col N=lane, K=0,1; lanes 16-31 K=2,3.
        const float gj = gs[(J << 4) + rl];
        v2f B;
        B.x = lo ? gj : (kb * gj * gj);  // K0 = g_j | K2 = b_j
        B.y = B1;                        // K1 = 1   | K3 = 0
        v8f cz = {0.f,0.f,0.f,0.f,0.f,0.f,0.f,0.f};
        v8f t = __builtin_amdgcn_wmma_f32_16x16x4_f32(
            /*neg_a=*/false, A, /*neg_b=*/false, B,
            /*c_mod=*/(short)0, cz, /*reuse_a=*/false, /*reuse_b=*/false);
        #pragma unroll
        for (int k = 0; k < 8; ++k)
            acc[k] += __builtin_amdgcn_exp2f(t[k]);   // v_exp_f32 (base-2)
    }

    // C/D layout: VGPR k holds row rbase+k (lanes 0-15) and rbase+k+8 (lanes 16-31),
    // striped across N=lane. Reduce over the 16 lanes of each half-wave.
    #pragma unroll
    for (int k = 0; k < 8; ++k) {
        float v = acc[k];
        v += __shfl_xor(v, 1, 32);
        v += __shfl_xor(v, 2, 32);
        v += __shfl_xor(v, 4, 32);
        v += __shfl_xor(v, 8, 32);
        acc[k] = v;
    }

    const float scale = 2.0f * nv * 0.69314718055994530942f;  // 2*nv*ln2
    if (lane == 0 || lane == 16) {
        const int roff = lo ? 0 : 8;
        #pragma unroll
        for (int k = 0; k < 8; ++k) {
            const int i = rbase + roff + k;
            const float d = pred[i] - gs[i];
            // loss = 2nv*ln(sum) + (p-g)^2  (diagonal folds to +(p-g)^2 exactly)
            out[i] = scale * __builtin_amdgcn_logf(acc[k]) + d * d;  // v_log_f32
        }
    }
}

extern "C" void kernel_launch(void* const* d_in, const int* in_sizes, int n_in,
                              void* d_out, int out_size, void* d_ws, size_t ws_size,
                              hipStream_t stream) {
    const float* pred  = (const float*)d_in[0];
    const float* gt    = (const float*)d_in[1];
    const float* sigma = (const float*)d_in[2];
    float* out = (float*)d_out;

    const int N = in_sizes[0];                 // 12288
    const int numRowTiles = N / 16;            // 768 waves of work
    const int blocks = (numRowTiles + WAVES_PER_BLOCK - 1) / WAVES_PER_BLOCK;

    hipLaunchKernelGGL(bmc_loss_wmma_kernel, dim3(blocks), dim3(BLOCK_THREADS),
                       0, stream, pred, gt, sigma, out, N);
}